// GFKWraper_7627861917944
// MI455X (gfx1250) — compile-verified
//
#include <hip/hip_runtime.h>
#include <stdint.h>

#define N_J        55
#define TILE_B     32
#define BLOCK      256
#define R_PER_B    495     // 55*9 floats of R per body
#define J_PER_B    165     // 55*3
#define V_PER_B    990     // 330*3 virtual output floats per body
#define OUT_PER_B  1155    // 385*3
#define VCHUNK_B   16      // virtual-staging chunk (bodies)

// SMPLX parents, constexpr so the unrolled chain folds every index.
constexpr int kParents[N_J] = {
  -1, 0, 0, 0, 1, 2, 3, 4, 5, 6, 7, 8, 9, 9, 9, 12, 13, 14, 16, 17, 18, 19,
  15, 15, 15, 20, 25, 26, 20, 28, 29, 20, 31, 32, 20, 34, 35, 20, 37, 38,
  21, 40, 41, 21, 43, 44, 21, 46, 47, 21, 49, 50, 21, 52, 53 };

typedef float v2f __attribute__((ext_vector_type(2)));
typedef float v8f __attribute__((ext_vector_type(8)));

// Pointer types matching the async-LDS builtin's parameters:
// arg0: v4i in global (as1), arg1: v4i in LDS (as3).
typedef int v4i __attribute__((vector_size(16)));
typedef __attribute__((address_space(1))) v4i gv4i;
typedef __attribute__((address_space(3))) v4i lv4i;

#if __has_builtin(__builtin_amdgcn_global_load_async_to_lds_b128)
#define HAVE_ASYNC_LDS 1
#endif

__device__ __forceinline__ void stage_b128(const float* g, float* l) {
#ifdef HAVE_ASYNC_LDS
  __builtin_amdgcn_global_load_async_to_lds_b128((gv4i*)g, (lv4i*)l, 0, 0);
#else
  *(float4*)l = *(const float4*)g;
#endif
}

__device__ __forceinline__ void stage_wait() {
#ifdef HAVE_ASYNC_LDS
#if __has_builtin(__builtin_amdgcn_s_wait_asynccnt)
  __builtin_amdgcn_s_wait_asynccnt(0);
#else
  asm volatile("s_wait_asynccnt 0" ::: "memory");
#endif
#endif
}

__global__ __launch_bounds__(BLOCK) void fk_virtual_kernel(
    const float* __restrict__ R, const float* __restrict__ J,
    float* __restrict__ out) {
  extern __shared__ float smem[];
  float* sR = smem;                       // 15840 floats (R tile)
  float* sJ = sR + TILE_B * R_PER_B;      //  5280 floats (J tile)
  float* sG = sJ + TILE_B * J_PER_B;      //  5280 floats (gpos)
  float* sV = sG + TILE_B * J_PER_B;      // 15840 floats (virtual staging)

  const int  t    = threadIdx.x;
  const int  lane = t & 31;
  const int  wave = t >> 5;
  const long tile = blockIdx.x;
  const long outTile = tile * TILE_B * (long)OUT_PER_B;

  // ---- Stage R and J tiles into LDS (CDNA5 async global->LDS, b128) ----
  const float* gR = R + tile * (TILE_B * R_PER_B);
  const float* gJ = J + tile * (TILE_B * J_PER_B);
  for (int i = t; i < TILE_B * R_PER_B / 4; i += BLOCK)
    stage_b128(gR + 4 * i, sR + 4 * i);
  for (int i = t; i < TILE_B * J_PER_B / 4; i += BLOCK)
    stage_b128(gJ + 4 * i, sJ + 4 * i);
  stage_wait();
  __syncthreads();

  // ---- Serial FK chain: wave 0, one lane per body (stride 495 -> bank-conflict free) ----
  if (wave == 0) {
    float*       g = sG + lane * J_PER_B;
    const float* j = sJ + lane * J_PER_B;
    const float* r = sR + lane * R_PER_B;
    g[0] = 0.f; g[1] = 0.f; g[2] = 0.f;
#pragma unroll
    for (int i = 1; i < N_J; ++i) {
      const int p = kParents[i];
      const float bx = j[i * 3 + 0] - j[p * 3 + 0];
      const float by = j[i * 3 + 1] - j[p * 3 + 1];
      const float bz = j[i * 3 + 2] - j[p * 3 + 2];
      const float* rp = r + p * 9;
      g[i * 3 + 0] = fmaf(rp[0], bx, fmaf(rp[1], by, fmaf(rp[2], bz, g[p * 3 + 0])));
      g[i * 3 + 1] = fmaf(rp[3], bx, fmaf(rp[4], by, fmaf(rp[5], bz, g[p * 3 + 1])));
      g[i * 3 + 2] = fmaf(rp[6], bx, fmaf(rp[7], by, fmaf(rp[8], bz, g[p * 3 + 2])));
    }
  }
  __syncthreads();

  // ---- Write gpos rows (flat floats 0..164 of each body): wave owns 4 bodies ----
  for (int b = wave * 4; b < wave * 4 + 4; ++b)
    for (int k = lane; k < J_PER_B; k += 32)
      out[outTile + b * (long)OUT_PER_B + k] = sG[b * J_PER_B + k];

  // ---- Virtual joints: rotated = A(16x4) x B(4x16 = BOX6^T, K padded to 4) ----
  // A row m = body m of the chunk; all 16 rows share one (joint n, component i).
  // B matrix is loop-invariant; build branch-free once.
  const bool lo  = (lane < 16);
  const int  col = lane & 15;
  const int  m   = lane & 15;          // A-matrix row = body index in chunk
  const int  M0  = lo ? 0 : 8;         // D row held in acc[0] for this lane half
  v2f bmat;
  bmat.x = lo ? ((col == 0) ? -0.05f : (col == 1) ? 0.05f : 0.f)   // K=0 row
              : ((col == 4) ? -0.05f : (col == 5) ? 0.05f : 0.f);  // K=2 row
  bmat.y = lo ? ((col == 2) ? -0.05f : (col == 3) ? 0.05f : 0.f)   // K=1 row
              : 0.f;                                               // K=3 pad

  for (int c = 0; c < TILE_B / VCHUNK_B; ++c) {
    const float* sRc = sR + c * (VCHUNK_B * R_PER_B);
    const float* sGc = sG + c * (VCHUNK_B * J_PER_B);
#if __has_builtin(__builtin_amdgcn_wmma_f32_16x16x4_f32)
    const float* aB = sRc + m * R_PER_B;    // per-lane A base (body m)
    const float* gB = sGc + M0 * J_PER_B;   // gpos base for D rows, stride 165/body
    float*       vB = sV + M0 * V_PER_B;    // sV base for D rows, stride 990/body
    const int tiles = N_J * 3;              // 165 (n,i) pairs; w = 3n+i
    for (int w = wave; w < tiles; w += BLOCK / 32) {  // wave-uniform: EXEC full at WMMA
      // A[m][k] = R[body m][n][i][k] = sRc[m*495 + 3w + k]  (n*9+3i == 3w)
      const float* r3 = aB + 3 * w;
      float r0 = r3[0], r1 = r3[1], r2v = r3[2];
      v2f a;                                // branch-free: no exec divergence
      a.x = lo ? r0 : r2v;                  // lanes 0-15: K=0 | 16-31: K=2
      a.y = lo ? r1 : 0.f;                  // lanes 0-15: K=1 | 16-31: K=3 pad
      v8f acc = {};
      acc = __builtin_amdgcn_wmma_f32_16x16x4_f32(false, a, false, bmat,
                                                  (short)0, acc, false, false);
      // D[M][col] = rotated_i for body M0+v; per-v strides are constants:
      //   gpos:  (M0+v)*165 + w          -> stride 165 floats
      //   sV:    (M0+v)*990 + 18n+i+3col -> stride 990 floats
      if (col < 6) {
        const int i  = w % 3;
        const int tb = 6 * w - 5 * i + 3 * col;   // 18n + i + 3*col
        const float* gp = gB + w;
        float*       vp = vB + tb;
        float gv_[8];
        // Batch the 8 LDS loads first so only ONE dscnt wait is needed
        // (LDS completes in order; interleaving load/store serializes on waits).
#pragma unroll
        for (int v = 0; v < 8; ++v)
          gv_[v] = gp[v * J_PER_B];
#pragma unroll
        for (int v = 0; v < 8; ++v)
          vp[v * V_PER_B] = acc[v] + gv_[v];
      }
    }
#else
    // Fallback: BOX6 is axis-aligned -> rotated_i = sign*0.05*R[i][axis]
    for (int idx = t; idx < VCHUNK_B * V_PER_B; idx += BLOCK) {
      int bl = idx / V_PER_B, r2 = idx - bl * V_PER_B;
      int n = r2 / 18, k = (r2 % 18) / 3, i = r2 % 3;
      int axis = k >> 1;
      float s = (k & 1) ? 0.05f : -0.05f;
      sV[idx] = sGc[bl * J_PER_B + n * 3 + i] +
                s * sRc[bl * R_PER_B + n * 9 + i * 3 + axis];
    }
#endif
    __syncthreads();
    // Coalesced writeout of this chunk's virtual rows: wave owns 2 bodies.
    const long vOut = outTile + (long)c * VCHUNK_B * OUT_PER_B + J_PER_B;
    for (int bl = wave * 2; bl < wave * 2 + 2; ++bl)
      for (int k = lane; k < V_PER_B; k += 32)
        out[vOut + bl * (long)OUT_PER_B + k] = sV[bl * V_PER_B + k];
    __syncthreads();
  }
}

extern "C" void kernel_launch(void* const* d_in, const int* in_sizes, int n_in,
                              void* d_out, int out_size, void* d_ws, size_t ws_size,
                              hipStream_t stream) {
  (void)n_in; (void)d_ws; (void)ws_size; (void)out_size;
  const float* R = (const float*)d_in[0];
  const float* J = (const float*)d_in[1];
  float* out = (float*)d_out;
  const int B    = in_sizes[0] / R_PER_B;   // 32768
  const int grid = B / TILE_B;              // 1024
  const size_t lds = (size_t)(TILE_B * R_PER_B + 2 * TILE_B * J_PER_B +
                              VCHUNK_B * V_PER_B) * sizeof(float);  // 168960 B
  fk_virtual_kernel<<<grid, BLOCK, lds, stream>>>(R, J, out);
}